// GNN_29807073034304
// MI455X (gfx1250) — compile-verified
//
#include <hip/hip_runtime.h>

#define NUSER 100000
#define NBOOK 50000
#define NEDGE 2000000
#define NLBL  500000
#define CCH   64
#define HCH   128

typedef __attribute__((ext_vector_type(2))) float v2f;
typedef __attribute__((ext_vector_type(8))) float v8f;

#if __has_builtin(__builtin_amdgcn_wmma_f32_16x16x4_f32)
#define HAVE_WMMA_F32 1
__device__ __forceinline__ v8f wmma4(v2f a, v2f b, v8f c) {
  // D = A(16x4,f32) * B(4x16,f32) + C(16x16,f32)
  return __builtin_amdgcn_wmma_f32_16x16x4_f32(false, a, false, b, (short)0, c, false, false);
}
#else
#define HAVE_WMMA_F32 0
#endif

// ---------------- user input projection: [NU,3] @ [3,64] + b ----------------
__global__ void k_proj_user(const float* __restrict__ x, const float* __restrict__ w,
                            const float* __restrict__ bias, float* __restrict__ out) {
  int i = blockIdx.x * blockDim.x + threadIdx.x;
  if (i >= NUSER * CCH) return;
  int node = i >> 6, c = i & 63;
  float acc = bias[c];
#pragma unroll
  for (int k = 0; k < 3; ++k) acc = fmaf(x[node * 3 + k], w[k * CCH + c], acc);
  out[i] = acc;
}

// ---------------- book input projection: [NB,384] @ [384,64] + b (WMMA) ----
// Weights staged in LDS as K-pairs: sW[kp*64+n] = {W[2kp][n], W[2kp+1][n]}
__global__ void __launch_bounds__(256)
k_proj_book(const float* __restrict__ x, const float* __restrict__ w,
            const float* __restrict__ bias, float* __restrict__ out) {
  int tid = threadIdx.x;
  int wave = (blockIdx.x * blockDim.x + tid) >> 5;
  int lane = tid & 31;
  int m = lane & 15, half = lane >> 4;
  int row0 = wave << 4;
  bool active = row0 < NBOOK;
#if HAVE_WMMA_F32
  __shared__ v2f sW[64 * CCH];  // one K=128 chunk: 64 k-pairs x 64 cols = 32KB
  const float* xr = x + (size_t)(row0 + m) * 384;
  v8f zero = {0.f, 0.f, 0.f, 0.f, 0.f, 0.f, 0.f, 0.f};
  v8f acc[4] = {zero, zero, zero, zero};
  for (int k0 = 0; k0 < 384; k0 += 128) {
    __syncthreads();  // protect LDS from overwrite while previous chunk in use
    for (int idx = tid; idx < 64 * CCH; idx += 256) {
      int n = idx & 63, kp = idx >> 6;
      int k = k0 + 2 * kp;
      v2f v;
      v.x = w[k * CCH + n];
      v.y = w[(k + 1) * CCH + n];
      sW[idx] = v;
    }
    __syncthreads();
    if (active) {
      for (int k = k0; k < k0 + 128; k += 4) {
        int kk = k + 2 * half;
        v2f a;
        a.x = xr[kk];
        a.y = xr[kk + 1];
        int lp = (kk - k0) >> 1;
#pragma unroll
        for (int t = 0; t < 4; ++t)
          acc[t] = wmma4(a, sW[lp * CCH + t * 16 + m], acc[t]);
      }
    }
  }
  if (active) {
#pragma unroll
    for (int t = 0; t < 4; ++t) {
      int n = t * 16 + m;
      float bv = bias[n];
#pragma unroll
      for (int r = 0; r < 8; ++r)
        out[(size_t)(row0 + r + 8 * half) * CCH + n] = acc[t][r] + bv;
    }
  }
#else
  if (!active) return;
  const float* xr = x + (size_t)(row0 + m) * 384;
  for (int n = half * 32; n < half * 32 + 32; ++n) {
    float acc = bias[n];
    for (int k = 0; k < 384; ++k) acc = fmaf(xr[k], w[k * CCH + n], acc);
    out[(size_t)(row0 + m) * CCH + n] = acc;
  }
#endif
}

// ---------------- zero a float buffer ----------------
__global__ void k_zero(float* __restrict__ p, int n) {
  int i = blockIdx.x * blockDim.x + threadIdx.x;
  if (i < n) p[i] = 0.0f;
}

// ---------------- in-degree counts ----------------
__global__ void k_count(const int* __restrict__ idx, float* __restrict__ cnt, int n) {
  int i = blockIdx.x * blockDim.x + threadIdx.x;
  if (i < n) atomicAdd(&cnt[idx[i]], 1.0f);
}

// ----- edge message scatter-add: agg[dst] += feat[src]; one wave per edge ---
__global__ void k_scatter(const int* __restrict__ gsrc, const int* __restrict__ gdst,
                          const float* __restrict__ feat, float* __restrict__ agg) {
  long long i = (long long)blockIdx.x * blockDim.x + threadIdx.x;
  if (i >= (long long)NEDGE * 32) return;
  int e = (int)(i >> 5), ch = ((int)i & 31) * 2;  // each thread: 2 channels
  int s = gsrc[e], d = gdst[e];
  const float2 v = *(const float2*)(feat + (size_t)s * CCH + ch);
  float* dst = agg + (size_t)d * CCH + ch;
  atomicAdd(dst, v.x);
  atomicAdd(dst + 1, v.y);
}

// --------- SAGE transform: out = act( (agg/max(cnt,1))@Wl + x@Wr + b ) ------
__global__ void __launch_bounds__(256)
k_sage(const float* __restrict__ agg, const float* __restrict__ cnt,
       const float* __restrict__ xdst,
       const float* __restrict__ wl, const float* __restrict__ wr,
       const float* __restrict__ bias, float* __restrict__ out,
       int n, int relu) {
  int tid = threadIdx.x;
  int wave = (blockIdx.x * blockDim.x + tid) >> 5;
  int lane = tid & 31;
  int m = lane & 15, half = lane >> 4;
  int row0 = wave << 4;
  bool active = row0 < n;
#if HAVE_WMMA_F32
  __shared__ v2f sWl[32 * CCH];  // 32 k-pairs x 64 cols = 16KB
  __shared__ v2f sWr[32 * CCH];  // 16KB
  for (int idx = tid; idx < 32 * CCH; idx += 256) {
    int nn = idx & 63, kp = idx >> 6;
    int k = 2 * kp;
    v2f a, b;
    a.x = wl[k * CCH + nn];       a.y = wl[(k + 1) * CCH + nn];
    b.x = wr[k * CCH + nn];       b.y = wr[(k + 1) * CCH + nn];
    sWl[idx] = a;
    sWr[idx] = b;
  }
  __syncthreads();
  if (!active) return;
  int row = row0 + m;
  float rc = 1.0f / fmaxf(cnt[row], 1.0f);
  const float* ar = agg + (size_t)row * CCH;
  const float* xr = xdst + (size_t)row * CCH;
  v8f zero = {0.f, 0.f, 0.f, 0.f, 0.f, 0.f, 0.f, 0.f};
  v8f acc[4] = {zero, zero, zero, zero};
  for (int k = 0; k < CCH; k += 4) {
    int kk = k + 2 * half;
    int lp = kk >> 1;
    v2f am, ax;
    am.x = ar[kk] * rc;  am.y = ar[kk + 1] * rc;
    ax.x = xr[kk];       ax.y = xr[kk + 1];
#pragma unroll
    for (int t = 0; t < 4; ++t) {
      int ncol = t * 16 + m;
      acc[t] = wmma4(am, sWl[lp * CCH + ncol], acc[t]);
      acc[t] = wmma4(ax, sWr[lp * CCH + ncol], acc[t]);
    }
  }
#pragma unroll
  for (int t = 0; t < 4; ++t) {
    int ncol = t * 16 + m;
    float bv = bias[ncol];
#pragma unroll
    for (int r = 0; r < 8; ++r) {
      float v = acc[t][r] + bv;
      if (relu) v = fmaxf(v, 0.0f);
      out[(size_t)(row0 + r + 8 * half) * CCH + ncol] = v;
    }
  }
#else
  if (!active) return;
  int row = row0 + m;
  float rc = 1.0f / fmaxf(cnt[row], 1.0f);
  const float* ar = agg + (size_t)row * CCH;
  const float* xr = xdst + (size_t)row * CCH;
  for (int ncol = half * 32; ncol < half * 32 + 32; ++ncol) {
    float v = bias[ncol];
    for (int k = 0; k < CCH; ++k)
      v = fmaf(ar[k] * rc, wl[k * CCH + ncol], fmaf(xr[k], wr[k * CCH + ncol], v));
    if (relu) v = fmaxf(v, 0.0f);
    out[(size_t)row * CCH + ncol] = v;
  }
#endif
}

// --- decoder: z=[u2[ls]|b2[ld]] (16x128) @ W1(128x128) +b1, relu, @W2 + b2 --
__global__ void __launch_bounds__(256)
k_decoder(const float* __restrict__ u2, const float* __restrict__ b2,
          const int* __restrict__ ls, const int* __restrict__ ld,
          const float* __restrict__ w1, const float* __restrict__ bias1,
          const float* __restrict__ w2, const float* __restrict__ bias2,
          float* __restrict__ out) {
  int tid = threadIdx.x;
  int wave = (blockIdx.x * blockDim.x + tid) >> 5;
  int lane = tid & 31;
  int m = lane & 15, half = lane >> 4;
  int row0 = wave << 4;
  bool active = row0 < NLBL;
  int row = active ? (row0 + m) : 0;
  const float* ur = u2 + (active ? (size_t)ls[row] * CCH : 0);
  const float* br = b2 + (active ? (size_t)ld[row] * CCH : 0);
#if HAVE_WMMA_F32
  __shared__ v2f sW[32 * HCH];  // one K=64 chunk: 32 k-pairs x 128 cols = 32KB
  v8f zero = {0.f, 0.f, 0.f, 0.f, 0.f, 0.f, 0.f, 0.f};
  v8f acc[8] = {zero, zero, zero, zero, zero, zero, zero, zero};
  for (int k0 = 0; k0 < HCH; k0 += 64) {
    __syncthreads();
    for (int idx = tid; idx < 32 * HCH; idx += 256) {
      int n = idx & 127, kp = idx >> 7;
      int k = k0 + 2 * kp;
      v2f v;
      v.x = w1[k * HCH + n];
      v.y = w1[(k + 1) * HCH + n];
      sW[idx] = v;
    }
    __syncthreads();
    if (active) {
      for (int k = k0; k < k0 + 64; k += 4) {
        int kk = k + 2 * half;  // even; kk,kk+1 on same side of the C=64 split
        const float* zsrc = (kk < CCH) ? (ur + kk) : (br + (kk - CCH));
        v2f a;
        a.x = zsrc[0];
        a.y = zsrc[1];
        int lp = (kk - k0) >> 1;
#pragma unroll
        for (int t = 0; t < 8; ++t)
          acc[t] = wmma4(a, sW[lp * HCH + t * 16 + m], acc[t]);
      }
    }
  }
  if (!active) return;
  float w2v[8], b1v[8];
#pragma unroll
  for (int t = 0; t < 8; ++t) {
    int ncol = t * 16 + m;
    w2v[t] = w2[ncol];
    b1v[t] = bias1[ncol];
  }
  float outb = bias2[0];
#pragma unroll
  for (int r = 0; r < 8; ++r) {
    float p = 0.0f;
#pragma unroll
    for (int t = 0; t < 8; ++t) {
      float hv = fmaxf(acc[t][r] + b1v[t], 0.0f);
      p = fmaf(hv, w2v[t], p);
    }
    // reduce the 16 lanes of each half (rows r / r+8)
    p += __shfl_xor(p, 1);
    p += __shfl_xor(p, 2);
    p += __shfl_xor(p, 4);
    p += __shfl_xor(p, 8);
    if (m == 0) out[row0 + r + 8 * half] = p + outb;
  }
#else
  if (active && half == 0) {
    float o = bias2[0];
    for (int n = 0; n < HCH; ++n) {
      float hv = bias1[n];
      for (int k = 0; k < CCH; ++k) hv = fmaf(ur[k], w1[k * HCH + n], hv);
      for (int k = 0; k < CCH; ++k) hv = fmaf(br[k], w1[(CCH + k) * HCH + n], hv);
      o = fmaf(fmaxf(hv, 0.0f), w2[n], o);
    }
    out[row] = o;
  }
#endif
}

extern "C" void kernel_launch(void* const* d_in, const int* in_sizes, int n_in,
                              void* d_out, int out_size, void* d_ws, size_t ws_size,
                              hipStream_t stream) {
  (void)in_sizes; (void)n_in; (void)out_size; (void)ws_size;
  const float* user_x     = (const float*)d_in[0];
  const float* book_x     = (const float*)d_in[1];
  const int*   edge_src   = (const int*)d_in[2];
  const int*   edge_dst   = (const int*)d_in[3];
  const int*   label_src  = (const int*)d_in[4];
  const int*   label_dst  = (const int*)d_in[5];
  const float* user_lin_w = (const float*)d_in[6];
  const float* user_lin_b = (const float*)d_in[7];
  const float* book_lin_w = (const float*)d_in[8];
  const float* book_lin_b = (const float*)d_in[9];
  const float* w1_ub_l = (const float*)d_in[10];
  const float* b1_ub   = (const float*)d_in[11];
  const float* w1_ub_r = (const float*)d_in[12];
  const float* w1_bu_l = (const float*)d_in[13];
  const float* b1_bu   = (const float*)d_in[14];
  const float* w1_bu_r = (const float*)d_in[15];
  const float* w2_ub_l = (const float*)d_in[16];
  const float* b2_ub   = (const float*)d_in[17];
  const float* w2_ub_r = (const float*)d_in[18];
  const float* w2_bu_l = (const float*)d_in[19];
  const float* b2_bu   = (const float*)d_in[20];
  const float* w2_bu_r = (const float*)d_in[21];
  const float* dec_w1  = (const float*)d_in[22];
  const float* dec_b1  = (const float*)d_in[23];
  const float* dec_w2  = (const float*)d_in[24];
  const float* dec_b2  = (const float*)d_in[25];
  float* out = (float*)d_out;

  // workspace layout (floats), ~103 MB total
  float* ws = (float*)d_ws;
  size_t off = 0;
  float* u0    = ws + off; off += (size_t)NUSER * CCH;  // 6.4M
  float* b0    = ws + off; off += (size_t)NBOOK * CCH;  // 3.2M
  float* u1    = ws + off; off += (size_t)NUSER * CCH;  // 6.4M
  float* b1buf = ws + off; off += (size_t)NBOOK * CCH;  // 3.2M
  float* agg   = ws + off; off += (size_t)NUSER * CCH;  // 6.4M (shared, max size)
  float* cnt_u = ws + off; off += NUSER;
  float* cnt_b = ws + off; off += NBOOK;
  float* u2 = u0;  // u0 dead after layer-1 user side
  float* b2 = b0;  // b0 dead after layer-1 user side

  const int BT = 256;
  const int sage_blk_b = (NBOOK / 16 * 32 + BT - 1) / BT;
  const int sage_blk_u = (NUSER / 16 * 32 + BT - 1) / BT;
  const int sc_blocks  = (int)(((long long)NEDGE * 32 + BT - 1) / BT);

  // input projections
  k_proj_user<<<(NUSER * CCH + BT - 1) / BT, BT, 0, stream>>>(user_x, user_lin_w, user_lin_b, u0);
  k_proj_book<<<(NBOOK / 16 * 32 + BT - 1) / BT, BT, 0, stream>>>(book_x, book_lin_w, book_lin_b, b0);

  // degree counts (same for both layers)
  k_zero<<<(NBOOK + BT - 1) / BT, BT, 0, stream>>>(cnt_b, NBOOK);
  k_zero<<<(NUSER + BT - 1) / BT, BT, 0, stream>>>(cnt_u, NUSER);
  k_count<<<(NEDGE + BT - 1) / BT, BT, 0, stream>>>(edge_dst, cnt_b, NEDGE);
  k_count<<<(NEDGE + BT - 1) / BT, BT, 0, stream>>>(edge_src, cnt_u, NEDGE);

  // ---- layer 1: book <- users ----
  k_zero<<<(NBOOK * CCH + BT - 1) / BT, BT, 0, stream>>>(agg, NBOOK * CCH);
  k_scatter<<<sc_blocks, BT, 0, stream>>>(edge_src, edge_dst, u0, agg);
  k_sage<<<sage_blk_b, BT, 0, stream>>>(agg, cnt_b, b0, w1_ub_l, w1_ub_r, b1_ub, b1buf, NBOOK, 1);
  // ---- layer 1: user <- books ----
  k_zero<<<(NUSER * CCH + BT - 1) / BT, BT, 0, stream>>>(agg, NUSER * CCH);
  k_scatter<<<sc_blocks, BT, 0, stream>>>(edge_dst, edge_src, b0, agg);
  k_sage<<<sage_blk_u, BT, 0, stream>>>(agg, cnt_u, u0, w1_bu_l, w1_bu_r, b1_bu, u1, NUSER, 1);

  // ---- layer 2: book <- users ----
  k_zero<<<(NBOOK * CCH + BT - 1) / BT, BT, 0, stream>>>(agg, NBOOK * CCH);
  k_scatter<<<sc_blocks, BT, 0, stream>>>(edge_src, edge_dst, u1, agg);
  k_sage<<<sage_blk_b, BT, 0, stream>>>(agg, cnt_b, b1buf, w2_ub_l, w2_ub_r, b2_ub, b2, NBOOK, 0);
  // ---- layer 2: user <- books ----
  k_zero<<<(NUSER * CCH + BT - 1) / BT, BT, 0, stream>>>(agg, NUSER * CCH);
  k_scatter<<<sc_blocks, BT, 0, stream>>>(edge_dst, edge_src, b1buf, agg);
  k_sage<<<sage_blk_u, BT, 0, stream>>>(agg, cnt_u, u1, w2_bu_l, w2_bu_r, b2_bu, u2, NUSER, 0);

  // ---- edge decoder ----
  k_decoder<<<(NLBL / 16 * 32 + BT - 1) / BT, BT, 0, stream>>>(
      u2, b2, label_src, label_dst, dec_w1, dec_b1, dec_w2, dec_b2, out);
}